// AttnAggregator1_20529943675140
// MI455X (gfx1250) — compile-verified
//
#include <hip/hip_runtime.h>
#include <math.h>

// MI455X / gfx1250 attention-aggregator.
// Algebra: masked-self => softmax over 32 neighbors only;
//   scores = n_s . (Q Wk^T) + bk.Q ;  out = (sum_s attn_s n_s) @ Wv + bv
// => three [16,256]x[256,256] f32 GEMMs per 16-item group via V_WMMA_F32_16X16X4_F32.
// Weights are prepacked (tiny kernel) into exact WMMA B-fragment order so the
// GEMM inner loop is: 1x ds_load_b64 (A) + 1x global_load_b64 (B, 256B/wave
// contiguous) + 1x v_wmma, with two independent accumulation chains.

typedef __attribute__((ext_vector_type(2))) float v2f;
typedef __attribute__((ext_vector_type(8))) float v8f;

#define BATCH  10000
#define NSAMP  32
#define FEAT   256
#define EMB    256
#define ITEMS  16            // items per workgroup
#define NTHR   512           // 16 waves of 32
#define KSTEPS 64            // 256 / 4
#define PACK_ELEMS (16 * KSTEPS * 32 * 2)  // 65536 floats = 256KB per matrix

__device__ __forceinline__ float wave_allreduce_add(float v) {
#pragma unroll
  for (int m = 16; m >= 1; m >>= 1) v += __shfl_xor(v, m, 32);
  return v;
}
__device__ __forceinline__ float wave_allreduce_max(float v) {
#pragma unroll
  for (int m = 16; m >= 1; m >>= 1) v = fmaxf(v, __shfl_xor(v, m, 32));
  return v;
}

// ---------------------------------------------------------------------------
// Prepack one 256x256 weight matrix into WMMA f32 16x16x4 B-fragment order:
//   Bp[((ntile*64 + kstep)*32 + lane)] = { B[k0][n], B[k0+1][n] } (float2)
// with n = ntile*16 + (lane&15), k0 = kstep*4 + (lane>>4)*2.
// BT=false: B[k][n] = W[k*256 + n]   (Wq, Wv)
// BT=true : B[k][n] = W[n*256 + k]   (Wk^T)
// ---------------------------------------------------------------------------
template <bool BT>
__global__ __launch_bounds__(512) void prepack_kernel(const float* __restrict__ W,
                                                      float* __restrict__ Bp) {
  const int t = blockIdx.x * 512 + threadIdx.x;   // 0 .. 32767
  const int lane  = t & 31;
  const int kstep = (t >> 5) & (KSTEPS - 1);
  const int ntile = t >> 11;
  const int n  = ntile * 16 + (lane & 15);
  const int k0 = kstep * 4 + ((lane >> 4) << 1);
  float b0, b1;
  if (BT) {
    b0 = W[(size_t)n * EMB + k0];
    b1 = W[(size_t)n * EMB + k0 + 1];
  } else {
    b0 = W[(size_t)k0 * EMB + n];
    b1 = W[(size_t)(k0 + 1) * EMB + n];
  }
  ((float2*)Bp)[t] = make_float2(b0, b1);
}

// ---------------------------------------------------------------------------
// One 16x16 output tile of A[16,256] @ B[256,256], K accumulated over 256.
// A in LDS (ds_load_b64 per frag). B either prepacked fragment-order (PACKED)
// or read directly from the row-major / transposed weight (fallback).
// Two independent WMMA accumulation chains (even/odd K-steps).
// ---------------------------------------------------------------------------
template <bool PACKED, bool BT>
__device__ __forceinline__ v8f gemm_tile16(const float (*__restrict__ A)[FEAT],
                                           const float* __restrict__ W,
                                           int ntile, int lane) {
  v8f acc0 = 0.0f, acc1 = 0.0f;
  const int row   = lane & 15;
  const int khalf = (lane >> 4) * 2;
  const int n     = ntile * 16 + (lane & 15);
  const float2* __restrict__ bp =
      (const float2*)W + (size_t)ntile * KSTEPS * 32 + lane;
#pragma unroll 8
  for (int ks = 0; ks < KSTEPS; ++ks) {
    const int k0 = ks * 4 + khalf;
    const float2 av = *(const float2*)&A[row][k0];
    const v2f a = {av.x, av.y};
    v2f b;
    if (PACKED) {
      const float2 bv2 = bp[(size_t)ks * 32];
      b = (v2f){bv2.x, bv2.y};
    } else if (BT) {
      const float2 bv2 = *(const float2*)&W[(size_t)n * EMB + k0];
      b = (v2f){bv2.x, bv2.y};
    } else {
      b = (v2f){W[(size_t)k0 * EMB + n], W[(size_t)(k0 + 1) * EMB + n]};
    }
    if (ks & 1)
      acc1 = __builtin_amdgcn_wmma_f32_16x16x4_f32(false, a, false, b,
                                                   (short)0, acc1, false, false);
    else
      acc0 = __builtin_amdgcn_wmma_f32_16x16x4_f32(false, a, false, b,
                                                   (short)0, acc0, false, false);
  }
  return acc0 + acc1;
}

template <bool PACKED>
__global__ __launch_bounds__(NTHR) void attn_agg_kernel(
    const int*   __restrict__ nodes,
    const int*   __restrict__ neigh_idx,
    const float* __restrict__ id2feat,
    const float* __restrict__ Bq,  const float* __restrict__ bq,
    const float* __restrict__ Bkt, const float* __restrict__ bk,
    const float* __restrict__ Bv,  const float* __restrict__ bv,
    float* __restrict__ out) {
  __shared__ float sA [ITEMS][FEAT];  // gathered self features
  __shared__ float sQ [ITEMS][EMB];   // Q = sA@Wq + bq
  __shared__ float sQp[ITEMS][FEAT];  // q' = Q @ Wk^T
  __shared__ float sH [ITEMS][FEAT];  // attn-weighted neighbor features

  const int tid  = threadIdx.x;
  const int wave = tid >> 5;          // 0..15 : N-tile for GEMMs, item for VALU phases
  const int lane = tid & 31;
  const int g    = blockIdx.x;        // 16-item group

  // ---- Phase 1: gather 16 self rows into LDS (coalesced along feature dim) ----
  for (int idx = tid; idx < ITEMS * FEAT; idx += NTHR) {
    const int r = idx >> 8, c = idx & 255;
    const int node = nodes[g * ITEMS + r];
    sA[r][c] = id2feat[(size_t)node * FEAT + c];
  }
  __syncthreads();

  // ---- Phase 2 (WMMA): Q = sA @ Wq + bq ; wave w owns emb cols [16w,16w+16) ----
  {
    const v8f acc = gemm_tile16<PACKED, false>(sA, Bq, wave, lane);
    const int n = wave * 16 + (lane & 15);
    const float bqn = bq[n];
#pragma unroll
    for (int v = 0; v < 8; ++v) {
      const int m = (lane >> 4) * 8 + v;   // C/D layout: lane<16 rows 0-7, else 8-15
      sQ[m][n] = acc[v] + bqn;
    }
  }
  __syncthreads();

  // ---- Phase 3 (WMMA): q' = Q @ Wk^T ----
  {
    const v8f acc = gemm_tile16<PACKED, true>(sQ, Bkt, wave, lane);
    const int n = wave * 16 + (lane & 15);
#pragma unroll
    for (int v = 0; v < 8; ++v) {
      const int m = (lane >> 4) * 8 + v;
      sQp[m][n] = acc[v];
    }
  }
  __syncthreads();

  // ---- Phase 4: scores + softmax. Wave w handles item w's 32 neighbors. ----
  const int item = g * ITEMS + wave;

  float bkq;  // bk . Q  (score bias term)
  {
    const float4 q0 = *(const float4*)&sQ[wave][lane * 8];
    const float4 q1 = *(const float4*)&sQ[wave][lane * 8 + 4];
    const float4 b0 = *(const float4*)&bk[lane * 8];
    const float4 b1 = *(const float4*)&bk[lane * 8 + 4];
    float p = q0.x * b0.x + q0.y * b0.y + q0.z * b0.z + q0.w * b0.w +
              q1.x * b1.x + q1.y * b1.y + q1.z * b1.z + q1.w * b1.w;
    bkq = wave_allreduce_add(p);
  }

  const float4 qp0 = *(const float4*)&sQp[wave][lane * 8];
  const float4 qp1 = *(const float4*)&sQp[wave][lane * 8 + 4];
  const int myidx = neigh_idx[(size_t)item * NSAMP + lane];  // lane s holds neighbor s

  float myscore = 0.0f;
  for (int s = 0; s < NSAMP; ++s) {
    const int nb = __shfl(myidx, s, 32);  // wave-uniform neighbor id
    const float4* rp = (const float4*)(id2feat + (size_t)nb * FEAT + lane * 8);
    const float4 r0 = rp[0];
    const float4 r1 = rp[1];
    const float p = r0.x * qp0.x + r0.y * qp0.y + r0.z * qp0.z + r0.w * qp0.w +
                    r1.x * qp1.x + r1.y * qp1.y + r1.z * qp1.z + r1.w * qp1.w;
    const float sc = wave_allreduce_add(p) + bkq;
    if (lane == s) myscore = sc;          // lane s keeps score s in a register
  }

  const float mx     = wave_allreduce_max(myscore);
  const float e      = __expf(myscore - mx);
  const float denom  = wave_allreduce_add(e);
  const float myattn = e / denom;         // lane s holds attn weight s

  // ---- Phase 5: h = sum_s attn_s * neigh_s ; lane covers 8 feature cols ----
  float hacc[8];
#pragma unroll
  for (int j = 0; j < 8; ++j) hacc[j] = 0.0f;
  for (int s = 0; s < NSAMP; ++s) {
    const float a  = __shfl(myattn, s, 32);
    const int   nb = __shfl(myidx, s, 32);
    const float4* rp = (const float4*)(id2feat + (size_t)nb * FEAT + lane * 8);
    const float4 r0 = rp[0];
    const float4 r1 = rp[1];
    hacc[0] += a * r0.x; hacc[1] += a * r0.y; hacc[2] += a * r0.z; hacc[3] += a * r0.w;
    hacc[4] += a * r1.x; hacc[5] += a * r1.y; hacc[6] += a * r1.z; hacc[7] += a * r1.w;
  }
#pragma unroll
  for (int j = 0; j < 8; ++j) sH[wave][lane * 8 + j] = hacc[j];
  __syncthreads();

  // ---- Phase 6 (WMMA): out = sH @ Wv + bv, stored straight to global ----
  {
    const v8f acc = gemm_tile16<PACKED, false>(sH, Bv, wave, lane);
    const int n = wave * 16 + (lane & 15);
    const float bvn = bv[n];
#pragma unroll
    for (int v = 0; v < 8; ++v) {
      const int m = (lane >> 4) * 8 + v;
      out[(size_t)(g * ITEMS + m) * EMB + n] = acc[v] + bvn;
    }
  }
}

extern "C" void kernel_launch(void* const* d_in, const int* in_sizes, int n_in,
                              void* d_out, int out_size, void* d_ws, size_t ws_size,
                              hipStream_t stream) {
  (void)in_sizes; (void)n_in; (void)out_size;
  const int*   nodes     = (const int*)  d_in[0];
  const int*   neigh_idx = (const int*)  d_in[1];
  const float* id2feat   = (const float*)d_in[2];
  const float* Wq        = (const float*)d_in[3];
  const float* bq        = (const float*)d_in[4];
  const float* Wk        = (const float*)d_in[5];
  const float* bk        = (const float*)d_in[6];
  const float* Wv        = (const float*)d_in[7];
  const float* bv        = (const float*)d_in[8];
  float* out = (float*)d_out;

  const dim3 grid(BATCH / ITEMS);  // 625 groups of 16 items
  const dim3 block(NTHR);          // 16 wave32 waves

  if (ws_size >= (size_t)3 * PACK_ELEMS * sizeof(float)) {
    float* ws = (float*)d_ws;
    float* pq = ws;
    float* pk = ws + PACK_ELEMS;
    float* pv = ws + 2 * PACK_ELEMS;
    prepack_kernel<false><<<64, 512, 0, stream>>>(Wq, pq);
    prepack_kernel<true ><<<64, 512, 0, stream>>>(Wk, pk);
    prepack_kernel<false><<<64, 512, 0, stream>>>(Wv, pv);
    attn_agg_kernel<true><<<grid, block, 0, stream>>>(
        nodes, neigh_idx, id2feat, pq, bq, pk, bk, pv, bv, out);
  } else {
    attn_agg_kernel<false><<<grid, block, 0, stream>>>(
        nodes, neigh_idx, id2feat, Wq, bq, Wk, bk, Wv, bv, out);
  }
}